// RiemannianAttentionBias_8615704396365
// MI455X (gfx1250) — compile-verified
//
#include <hip/hip_runtime.h>
#include <math.h>

typedef __attribute__((ext_vector_type(2))) float v2f;
typedef __attribute__((ext_vector_type(8))) float v8f;

#define EPS_F    1e-5f
#define LDM      65      // padded LDS stride for 64x64 matrices (conflict-free col reads)
#define LDX      130     // padded LDS stride for 64x128 x tile
#define NSQRT    5       // # of Newton-Schulz square roots (spectrum -> [eps^(1/32),1])
#define NITER    7       // NS iterations per square root
#define NTERM    8       // Mercator series terms for log(I+E)
#define MAT_F    (64*LDM)            // floats per padded 64x64 matrix = 4160
#define SMEM_F   (4*MAT_F)           // 4 matrix buffers; x tile overlaps buffers 0+1
#define SMEM_B   (SMEM_F*sizeof(float))

__device__ __forceinline__ v8f wmma_f32(v2f a, v2f b, v8f c) {
  // V_WMMA_F32_16X16X4_F32: D = A(16x4) * B(4x16) + C(16x16), fp32 throughout.
  return __builtin_amdgcn_wmma_f32_16x16x4_f32(false, a, false, b, (short)0, c,
                                               false, false);
}

// Store one 16x16 f32 C/D tile with a fused epilogue.
// NSMODE: C = 0.5*(3*I - acc)      (Newton-Schulz T-step)
// else  : C = s*acc + d*I          (plain store: s=1,d=0; diag-add; final scale)
template <bool NSMODE>
__device__ __forceinline__ void store_tile(float* __restrict__ C, int mbase, int nc,
                                           v8f acc, float s, float d) {
  #pragma unroll
  for (int r = 0; r < 8; ++r) {
    const bool diag = (mbase + r) == nc;
    float v = acc[r];
    if (NSMODE) v = 0.5f * ((diag ? 3.0f : 0.0f) - v);
    else        v = s * v + (diag ? d : 0.0f);
    C[(mbase + r) * LDM + nc] = v;
  }
}

// C(64x64) = epilogue(A(64x64) * B(64x64)); row-major stride LDM in LDS.
// 4 waves: wave w owns rows [16w,16w+16). K-outer / N-inner with 4 independent
// accumulators -> 4 concurrent WMMA chains, A fragment loaded once per K-step.
template <bool NSMODE>
__device__ __forceinline__ void block_mm64(const float* __restrict__ A,
                                           const float* __restrict__ B,
                                           float* __restrict__ C,
                                           float s, float d) {
  const int lane = threadIdx.x & 31;
  const int w    = threadIdx.x >> 5;
  const int half = lane >> 4;          // lanes 16-31 hold the +2 K pair / +8 M rows
  const int l15  = lane & 15;
  const int mr   = w * 16 + l15;       // A row owned by this lane
  v8f a0 = {}, a1 = {}, a2 = {}, a3 = {};
  #pragma unroll
  for (int k0 = 0; k0 < 64; k0 += 4) {
    const int ka = k0 + 2 * half;
    v2f a;  a.x  = A[mr * LDM + ka];            a.y  = A[mr * LDM + ka + 1];
    v2f b0; b0.x = B[ka * LDM + l15];           b0.y = B[(ka + 1) * LDM + l15];
    v2f b1; b1.x = B[ka * LDM + 16 + l15];      b1.y = B[(ka + 1) * LDM + 16 + l15];
    v2f b2; b2.x = B[ka * LDM + 32 + l15];      b2.y = B[(ka + 1) * LDM + 32 + l15];
    v2f b3; b3.x = B[ka * LDM + 48 + l15];      b3.y = B[(ka + 1) * LDM + 48 + l15];
    a0 = wmma_f32(a, b0, a0);
    a1 = wmma_f32(a, b1, a1);
    a2 = wmma_f32(a, b2, a2);
    a3 = wmma_f32(a, b3, a3);
  }
  const int mbase = w * 16 + 8 * half;
  store_tile<NSMODE>(C, mbase,      l15, a0, s, d);
  store_tile<NSMODE>(C, mbase, 16 + l15, a1, s, d);
  store_tile<NSMODE>(C, mbase, 32 + l15, a2, s, d);
  store_tile<NSMODE>(C, mbase, 48 + l15, a3, s, d);
}

__global__ void __launch_bounds__(128)
riem_bias_kernel(const float* __restrict__ x, const float* __restrict__ hs,
                 float* __restrict__ out, int BN) {
  extern __shared__ float smem[];
  float* A0 = smem;
  float* A1 = smem + MAT_F;
  float* A2 = smem + 2 * MAT_F;
  float* A3 = smem + 3 * MAT_F;
  float* xs = smem;                    // 64x130 x tile overlaps A0+A1 (dies after cov)
  __shared__ float red[1];

  const int b    = blockIdx.x;
  const int tid  = threadIdx.x;
  const int lane = tid & 31;
  const int w    = tid >> 5;
  const int half = lane >> 4;
  const int l15  = lane & 15;
  if (b >= BN) return;

  // ---- 1) stage x[b] (64x128 fp32) into LDS, padded rows ----
  const float* xg = x + (size_t)b * (64 * 128);
  for (int e = tid * 4; e < 64 * 128; e += 128 * 4) {
    float4 v = *reinterpret_cast<const float4*>(xg + e);
    const int r = e >> 7, c = e & 127;
    xs[r * LDX + c + 0] = v.x;  xs[r * LDX + c + 1] = v.y;
    xs[r * LDX + c + 2] = v.z;  xs[r * LDX + c + 3] = v.w;
  }
  __syncthreads();

  // ---- 2) S = xs * xs^T / 128 + eps*I -> A2 (fp32 WMMA, K=128, 4 acc chains) ----
  {
    const int   mr   = w * 16 + l15;
    const float invD = 1.0f / 128.0f;
    v8f a0 = {}, a1 = {}, a2 = {}, a3 = {};
    #pragma unroll
    for (int k0 = 0; k0 < 128; k0 += 4) {
      const int ka = k0 + 2 * half;
      v2f a;  a.x  = xs[mr * LDX + ka];             a.y  = xs[mr * LDX + ka + 1];
      v2f b0; b0.x = xs[(l15     ) * LDX + ka];     b0.y = xs[(l15     ) * LDX + ka + 1];
      v2f b1; b1.x = xs[(16 + l15) * LDX + ka];     b1.y = xs[(16 + l15) * LDX + ka + 1];
      v2f b2; b2.x = xs[(32 + l15) * LDX + ka];     b2.y = xs[(32 + l15) * LDX + ka + 1];
      v2f b3; b3.x = xs[(48 + l15) * LDX + ka];     b3.y = xs[(48 + l15) * LDX + ka + 1];
      a0 = wmma_f32(a, b0, a0);
      a1 = wmma_f32(a, b1, a1);
      a2 = wmma_f32(a, b2, a2);
      a3 = wmma_f32(a, b3, a3);
    }
    const int mbase = w * 16 + 8 * half;
    store_tile<false>(A2, mbase,      l15, a0, invD, EPS_F);
    store_tile<false>(A2, mbase, 16 + l15, a1, invD, EPS_F);
    store_tile<false>(A2, mbase, 32 + l15, a2, invD, EPS_F);
    store_tile<false>(A2, mbase, 48 + l15, a3, invD, EPS_F);
  }
  __syncthreads();

  // ---- 3) trace normalization: A = S/tr(S);  log(S) = log(tr)*I + log(A) ----
  if (tid == 0) red[0] = 0.0f;
  __syncthreads();
  if (tid < 64) atomicAdd(&red[0], A2[tid * LDM + tid]);   // ds_add_f32
  __syncthreads();
  const float tr    = red[0];
  const float invtr = 1.0f / tr;
  const float logtr = logf(tr);
  __syncthreads();

  float *pY = A0, *pZ = A1, *pT = A2, *pW = A3;
  for (int e = tid; e < 4096; e += 128) {                  // Y = S/tr, Z = I
    const int r = e >> 6, c = e & 63;
    pY[r * LDM + c] = pT[r * LDM + c] * invtr;
    pZ[r * LDM + c] = (r == c) ? 1.0f : 0.0f;
  }
  __syncthreads();

  // ---- 4) NSQRT repeated square roots via coupled Newton-Schulz ----
  for (int sq = 0; sq < NSQRT; ++sq) {
    if (sq > 0) {
      for (int e = tid; e < 4096; e += 128) {              // reset Z = I
        const int r = e >> 6, c = e & 63;
        pZ[r * LDM + c] = (r == c) ? 1.0f : 0.0f;
      }
      __syncthreads();
    }
    for (int it = 0; it < NITER; ++it) {
      block_mm64<true >(pZ, pY, pT, 1.0f, 0.0f);           // T = (3I - Z*Y)/2 (fused)
      __syncthreads();
      block_mm64<false>(pY, pT, pW, 1.0f, 0.0f);           // W = Y*T   (new Y)
      __syncthreads();
      block_mm64<false>(pT, pZ, pY, 1.0f, 0.0f);           // Ybuf = T*Z (new Z)
      __syncthreads();
      float *oY = pY, *oZ = pZ, *oW = pW;
      pY = oW; pZ = oY; pW = oZ;                           // rotate buffers
    }
  }

  // ---- 5) log(A^(1/2^k)) = log(I+E), E = Y - I, fused-Horner matrix series ----
  for (int e = tid; e < 4096; e += 128) {
    const int r = e >> 6, c = e & 63;
    if (r == c) pY[r * LDM + c] -= 1.0f;                   // E in place
  }
  __syncthreads();
  {
    const float cN   = ((NTERM & 1) ? 1.0f : -1.0f) / (float)NTERM;
    const float cNm1 = (((NTERM - 1) & 1) ? 1.0f : -1.0f) / (float)(NTERM - 1);
    for (int e = tid; e < 4096; e += 128) {                // H = cN*E + c_{N-1}*I
      const int r = e >> 6, c = e & 63;
      float v = cN * pY[r * LDM + c];
      if (r == c) v += cNm1;
      pT[r * LDM + c] = v;
    }
  }
  __syncthreads();
  for (int n = NTERM - 2; n >= 1; --n) {
    const float cn = ((n & 1) ? 1.0f : -1.0f) / (float)n;
    block_mm64<false>(pY, pT, pW, 1.0f, cn);               // W = E*H + c_n*I (fused)
    __syncthreads();
    float* t = pT; pT = pW; pW = t;
  }
  const float p2k = (float)(1 << NSQRT);
  block_mm64<false>(pY, pT, pW, p2k, logtr);               // L = 2^k*(E*H1) + log(tr)*I
  __syncthreads();

  // ---- 6) broadcast over heads: out[b,h,:,:] = L * hs[h]  (float4 streams) ----
  float hsr[8];
  #pragma unroll
  for (int h = 0; h < 8; ++h) hsr[h] = hs[h];
  float* outb = out + (size_t)b * 8 * 4096;
  for (int e4 = tid; e4 < 1024; e4 += 128) {
    const int e = e4 * 4;
    const int r = e >> 6, c = e & 63;
    float4 v;
    v.x = pW[r * LDM + c + 0]; v.y = pW[r * LDM + c + 1];
    v.z = pW[r * LDM + c + 2]; v.w = pW[r * LDM + c + 3];
    #pragma unroll
    for (int h = 0; h < 8; ++h) {
      float4 o;
      o.x = v.x * hsr[h]; o.y = v.y * hsr[h];
      o.z = v.z * hsr[h]; o.w = v.w * hsr[h];
      *reinterpret_cast<float4*>(outb + h * 4096 + e) = o;
    }
  }
}

extern "C" void kernel_launch(void* const* d_in, const int* in_sizes, int n_in,
                              void* d_out, int out_size, void* d_ws, size_t ws_size,
                              hipStream_t stream) {
  const float* x  = (const float*)d_in[0];   // (BN, 64, 128) fp32
  const float* hs = (const float*)d_in[1];   // (8,) fp32
  float* out = (float*)d_out;                // (BN, 8, 64, 64) fp32
  const int BN = in_sizes[0] / (64 * 128);

  // 65 KB dynamic LDS per block (gfx1250 WGP has 320 KB -> ~4 blocks/WGP).
  (void)hipFuncSetAttribute((const void*)riem_bias_kernel,
                            hipFuncAttributeMaxDynamicSharedMemorySize,
                            (int)SMEM_B);
  riem_bias_kernel<<<dim3(BN), dim3(128), SMEM_B, stream>>>(x, hs, out, BN);
}